// ModelSGLang_67293547594180
// MI455X (gfx1250) — compile-verified
//
#include <hip/hip_runtime.h>
#include <hip/hip_bf16.h>

// ---------------------------------------------------------------------------
// moe_align_block_size for MI455X (gfx1250, wave32).
// Deterministic stable counting sort: per-wave-chunk histogram -> per-expert
// chunk scan -> ballot-ranked scatter. Async global->LDS staging (ASYNCcnt),
// wave32 ballots, LDS atomics, global prefetch.
// ---------------------------------------------------------------------------

#define NE        64            // num_experts (reference constant)
#define EE        (NE + 1)      // kernel is called with E+1 experts
#define BS        128           // block_size (reference constant)
#define CHUNK     512           // tokens per wave-chunk
#define WAVES_PB  8             // waves per block (256 threads, wave32)

#if __has_builtin(__builtin_amdgcn_ballot_w32)
#define BALLOT32(x) __builtin_amdgcn_ballot_w32(x)
#else
#define BALLOT32(x) ((unsigned)__ballot(x))
#endif

#if defined(__AMDGCN__) && \
    __has_builtin(__builtin_amdgcn_global_load_async_to_lds_b128) && \
    __has_builtin(__builtin_amdgcn_s_wait_asynccnt)
#define HAVE_ASYNC 1
typedef int v4i_ __attribute__((__vector_size__(16)));
typedef __attribute__((address_space(1))) v4i_ g_v4i;   // global 16B vector
typedef __attribute__((address_space(3))) v4i_ l_v4i;   // LDS 16B vector
#else
#define HAVE_ASYNC 0
#endif

// ---------------------------------------------------------------------------
// K1: per-wave-chunk histogram. cnt layout is expert-major: cnt[e*C + chunk].
// ---------------------------------------------------------------------------
__global__ void moe_hist_kernel(const int* __restrict__ flat,
                                int* __restrict__ cnt, int T, int C) {
    __shared__ int hist[WAVES_PB][EE];
    const int w    = threadIdx.x >> 5;
    const int lane = threadIdx.x & 31;

    for (int e = lane; e < EE; e += 32) hist[w][e] = 0;
    __syncthreads();

    const int c    = blockIdx.x * WAVES_PB + w;
    const int base = c * CHUNK;
    const int4* f4 = reinterpret_cast<const int4*>(flat);

#pragma unroll
    for (int i = 0; i < CHUNK / (32 * 4); ++i) {
        const int idx4 = (base >> 2) + i * 32 + lane;
        __builtin_prefetch(f4 + idx4 + 64, 0, 1);   // global_prefetch_b8
        if (idx4 * 4 + 3 < T) {
            const int4 v = f4[idx4];
            atomicAdd(&hist[w][v.x], 1);
            atomicAdd(&hist[w][v.y], 1);
            atomicAdd(&hist[w][v.z], 1);
            atomicAdd(&hist[w][v.w], 1);
        } else {
            for (int k = 0; k < 4; ++k) {
                const int idx = idx4 * 4 + k;
                if (idx < T) atomicAdd(&hist[w][flat[idx]], 1);
            }
        }
    }
    __syncthreads();

    if (c < C)
        for (int e = lane; e < EE; e += 32) cnt[e * C + c] = hist[w][e];
}

// ---------------------------------------------------------------------------
// K2: one block per expert. In-place exclusive scan of cnt[e][0..C) over
// chunks; totals[e] = per-expert token count.
// ---------------------------------------------------------------------------
__global__ void moe_scan_kernel(int* __restrict__ cnt,
                                int* __restrict__ totals, int C) {
    __shared__ int s[256];
    __shared__ int carry;
    const int e = blockIdx.x;
    const int t = threadIdx.x;
    if (t == 0) carry = 0;
    __syncthreads();

    const int base = e * C;
    for (int tile = 0; tile * 256 < C; ++tile) {
        const int i = tile * 256 + t;
        const int v = (i < C) ? cnt[base + i] : 0;
        s[t] = v;
        __syncthreads();
        for (int off = 1; off < 256; off <<= 1) {
            const int x = (t >= off) ? s[t - off] : 0;
            __syncthreads();
            s[t] += x;
            __syncthreads();
        }
        const int excl = s[t] - v;
        const int c0   = carry;
        __syncthreads();
        if (i < C) cnt[base + i] = c0 + excl;
        if (t == 255) carry = c0 + s[255];
        __syncthreads();
    }
    if (t == 0) totals[e] = carry;
}

// ---------------------------------------------------------------------------
// K3: 65-wide padded cumsum; writes num_tokens_post_pad.
// ---------------------------------------------------------------------------
__global__ void moe_finalize_kernel(const int* __restrict__ totals,
                                    int* __restrict__ off_pad,
                                    int* __restrict__ cum_pad,
                                    int* __restrict__ ntp_out) {
    if (threadIdx.x == 0) {
        int run = 0;
        for (int e = 0; e < EE; ++e) {
            const int cv  = totals[e];
            const int pad = ((cv + BS - 1) / BS) * BS;
            off_pad[e] = run;
            run += pad;
            cum_pad[e] = run;
        }
        *ntp_out = run;
    }
}

// ---------------------------------------------------------------------------
// K4: fill only the padding slots (<= EE*(BS-1) per-expert + tail) with T.
// Blocks 0..EE-1: per-expert gap. Blocks EE..: tail [cum_total, M).
// ---------------------------------------------------------------------------
__global__ void moe_padfill_kernel(const int* __restrict__ totals,
                                   const int* __restrict__ off_pad,
                                   const int* __restrict__ cum_pad,
                                   int* __restrict__ sorted, int T, int M) {
    const int b = blockIdx.x;
    if (b < EE) {
        const int s0 = off_pad[b] + totals[b];
        const int e0 = cum_pad[b];
        for (int i = s0 + threadIdx.x; i < e0; i += blockDim.x) sorted[i] = T;
    } else {
        const int total  = cum_pad[EE - 1];
        const int nb2    = gridDim.x - EE;
        const int stride = nb2 * blockDim.x;
        for (int i = total + (b - EE) * blockDim.x + threadIdx.x; i < M;
             i += stride)
            sorted[i] = T;
    }
}

// ---------------------------------------------------------------------------
// K5: expert id per output block = searchsorted(cum_pad, bstart, 'right'),
// zero beyond padded total. cum_pad cached in LDS; 7-step binary search.
// ---------------------------------------------------------------------------
__global__ void moe_expertids_kernel(const int* __restrict__ cum_pad,
                                     int* __restrict__ expert_ids, int NB) {
    __shared__ int cp[EE];
    if (threadIdx.x < EE) cp[threadIdx.x] = cum_pad[threadIdx.x];
    __syncthreads();
    const int i = blockIdx.x * blockDim.x + threadIdx.x;
    if (i >= NB) return;
    const int bstart = i * BS;
    int eid = 0;
    if (bstart < cp[EE - 1]) {
        int lo = 0, hi = EE - 1;          // first idx with cp[idx] > bstart
        while (lo < hi) {
            const int mid = (lo + hi) >> 1;
            if (cp[mid] > bstart) hi = mid; else lo = mid + 1;
        }
        eid = lo;
    }
    expert_ids[i] = eid;
}

// ---------------------------------------------------------------------------
// K6: stable scatter. Each wave owns one 512-token chunk staged via async
// global->LDS (overlapped with 65 offset loads). Rank within a 32-token
// group via wave32 ballot bit-votes; cross-group stability via per-wave LDS
// running counters seeded with off_pad[e] + chunkOff[e][c].
// ---------------------------------------------------------------------------
__global__ void moe_scatter_kernel(const int* __restrict__ flat,
                                   const int* __restrict__ chunk_off,
                                   const int* __restrict__ off_pad,
                                   int* __restrict__ sorted, int T, int C) {
    __shared__ int running[WAVES_PB][72];
    __shared__ int stage[WAVES_PB][CHUNK];
    const int w    = threadIdx.x >> 5;
    const int lane = threadIdx.x & 31;
    const int c    = blockIdx.x * WAVES_PB + w;
    const bool haveChunk = (c < C);
    const int tokensBase = c * CHUNK;
    const bool full = haveChunk && (tokensBase + CHUNK <= T);

#if HAVE_ASYNC
    if (full) {
#pragma unroll
        for (int i = 0; i < CHUNK / (32 * 4); ++i) {
            g_v4i* g = (g_v4i*)(flat + tokensBase + i * 128 + lane * 4);
            l_v4i* l = (l_v4i*)&stage[w][i * 128 + lane * 4];
            __builtin_amdgcn_global_load_async_to_lds_b128(g, l, 0, 0);
        }
    }
#endif
    if (haveChunk) {
        for (int e = lane; e < EE; e += 32)
            running[w][e] = off_pad[e] + chunk_off[e * C + c];
    }
#if HAVE_ASYNC
    __builtin_amdgcn_s_wait_asynccnt(0);
#endif
    if (!haveChunk) return;

#pragma unroll 4
    for (int i = 0; i < CHUNK / 32; ++i) {
        const int idxLocal = i * 32 + lane;
        const int gidx     = tokensBase + idxLocal;
        const bool valid   = (gidx < T);
        int e;
#if HAVE_ASYNC
        if (full) e = stage[w][idxLocal];
        else      e = valid ? flat[gidx] : 128;
#else
        e = valid ? flat[gidx] : 128;
#endif
        if (!valid) e = 128;   // groups invalid lanes apart from any expert

        unsigned mask = 0xffffffffu;     // full waves: EXEC all-ones here
#pragma unroll
        for (int b = 0; b < 8; ++b) {
            const unsigned vote = BALLOT32((e >> b) & 1);
            mask &= ((e >> b) & 1) ? vote : ~vote;
        }
        const unsigned lt = mask & ((1u << lane) - 1u);
        const int rank    = __popc(lt);
        const int ei      = (e < EE) ? e : (EE - 1);   // harmless dummy slot
        const int base    = running[w][ei];            // ds_load (all lanes)
        if ((lt == 0u) && valid)                       // group leader updates
            running[w][ei] = base + __popc(mask);
        if (valid) sorted[base + rank] = gidx;         // stable position
    }
}

// ---------------------------------------------------------------------------
// Host launcher.  d_out (int32): [ sorted_token_ids(M) | expert_ids(NB) | 1 ]
// d_ws: [ cnt/chunk_off (EE*C) | totals(EE) | off_pad(EE) | cum_pad(EE) ]
// ---------------------------------------------------------------------------
extern "C" void kernel_launch(void* const* d_in, const int* in_sizes, int n_in,
                              void* d_out, int out_size, void* d_ws,
                              size_t ws_size, hipStream_t stream) {
    const int* flat = (const int*)d_in[0];
    const int  T    = in_sizes[0];               // NUM_TOKENS * TOP_K
    const int  M    = T + EE * (BS - 1);         // max_num_tokens_padded
    const int  NB   = M / BS;                    // max_num_m_blocks
    const int  C    = (T + CHUNK - 1) / CHUNK;   // wave-chunks

    int* ws       = (int*)d_ws;
    int* cnt      = ws;                          // EE*C, becomes chunk_off
    int* totals   = cnt + EE * C;
    int* off_pad  = totals + EE;
    int* cum_pad  = off_pad + EE;

    int* out        = (int*)d_out;
    int* sorted     = out;
    int* expert_ids = out + M;
    int* ntp        = out + M + NB;

    const int hb = (C + WAVES_PB - 1) / WAVES_PB;

    moe_hist_kernel<<<hb, 256, 0, stream>>>(flat, cnt, T, C);
    moe_scan_kernel<<<EE, 256, 0, stream>>>(cnt, totals, C);
    moe_finalize_kernel<<<1, 32, 0, stream>>>(totals, off_pad, cum_pad, ntp);
    moe_padfill_kernel<<<EE + 64, 128, 0, stream>>>(totals, off_pad, cum_pad,
                                                    sorted, T, M);
    moe_expertids_kernel<<<(NB + 255) / 256, 256, 0, stream>>>(cum_pad,
                                                               expert_ids, NB);
    moe_scatter_kernel<<<hb, 256, 0, stream>>>(flat, cnt, off_pad, sorted, T,
                                               C);
}